// KVCache_9328668967404
// MI455X (gfx1250) — compile-verified
//
#include <hip/hip_runtime.h>
#include <hip/hip_bf16.h>

// 16-byte vector (8 x bf16) as a clang ext-vector so the nontemporal
// builtins are legal and lower to global_load_b128/global_store_b128 with
// TH=NT on gfx1250.
typedef unsigned int v4u __attribute__((ext_vector_type(4)));

#define LAYERS     8
#define BSZ        4
#define MAX_SEQ    4096
#define KV_HEADS   8
#define HEAD_DIM   128
#define INSERT_LEN 16

// counts in 16-byte vectors
#define VPC   (LAYERS * BSZ * MAX_SEQ * KV_HEADS * HEAD_DIM / 8)  // 1<<24 per cache
#define VPL   (VPC / LAYERS)                                      // 1<<21 per layer
#define VPS   (KV_HEADS * HEAD_DIM / 8)                           // 128 per seq pos
#define VPROW (HEAD_DIM / 8)                                      // 16 per head row

#define VPT   4   // vectors per thread in the copy kernel (4 loads in flight)

// Kernel 1: stream-copy both caches input -> output, overwriting the
// [layer_idx, :, cur_pos:cur_pos+INSERT_LEN] slice with xk/xv (fused insert).
// Each thread moves 4 x 16B with disjoint-bitfield addressing:
//   v = group<<10 | k<<8 | lane   (lane = tid&255, k = 0..3)
// so every global_load/store_b128 is a fully coalesced 512B wave access and
// layer/b/r never change across k. Grid exactly covers 2*VPC vectors.
__global__ __launch_bounds__(256) void kv_cache_copy_insert(
    const v4u* __restrict__ kin, const v4u* __restrict__ vin,
    const v4u* __restrict__ xk,  const v4u* __restrict__ xv,
    const int* __restrict__ p_layer, const int* __restrict__ p_pos,
    v4u* __restrict__ kout, v4u* __restrict__ vout)
{
    const unsigned layer_idx = (unsigned)p_layer[0];   // uniform -> s_load_b32
    const unsigned cur_pos   = (unsigned)p_pos[0];

    const unsigned tid = blockIdx.x * 256u + threadIdx.x;   // [0, 2*VPC/VPT)
    const unsigned isv   = tid >> 22;                       // 0 = K, 1 = V
    const unsigned t     = tid & ((VPC / VPT) - 1u);        // [0, 1<<22)
    const unsigned group = t >> 8;                          // bits 10..23 of v
    const unsigned lane  = t & 255u;                        // bits 0..7 of v

    const v4u* __restrict__ cin  = isv ? vin : kin;
    const v4u* __restrict__ xin  = isv ? xv : xk;
    v4u*       __restrict__ cout = isv ? vout : kout;

    // k-invariant fields
    const unsigned vbase = (group << 10) | lane;
    const unsigned layer = vbase >> 21;
    const unsigned b     = (vbase >> 19) & (BSZ - 1u);
    const unsigned r     = vbase & (VPS - 1u);              // h*16 + d/8
    const bool     lhit  = (layer == layer_idx);

    v4u d[VPT];
#pragma unroll
    for (unsigned k = 0; k < VPT; ++k) {
        const unsigned v    = vbase | (k << 8);
        const unsigned s    = (v >> 7) & (MAX_SEQ - 1u);
        const unsigned srel = s - cur_pos;                  // wraps if s < cur_pos
        const v4u* src;
        if (lhit && srel < INSERT_LEN) {
            // xk/xv layout: [BSZ, INSERT_LEN, KV_HEADS, HEAD_DIM] -> 128 vecs/(b,l)
            src = xin + ((b * INSERT_LEN + srel) << 7) + r;
        } else {
            src = cin + v;
        }
        d[k] = __builtin_nontemporal_load(src);             // 4 loads in flight
    }
#pragma unroll
    for (unsigned k = 0; k < VPT; ++k) {
        __builtin_nontemporal_store(d[k], cout + (vbase | (k << 8)));
    }
}

// Kernel 2: GQA expansion of the updated layer slab.
// keys[b, s, h*n_rep + rr, d] = k_cache_out[layer_idx, b, s, h, d]
// One 16B load, n_rep 16B stores per thread. Grid covers 2*VPL vectors.
__global__ __launch_bounds__(256) void kv_gqa_expand(
    const v4u* __restrict__ kco, const v4u* __restrict__ vco,
    v4u* __restrict__ keys, v4u* __restrict__ vals,
    const int* __restrict__ p_layer, int n_rep)
{
    const unsigned layer_idx = (unsigned)p_layer[0];

    const unsigned tid = blockIdx.x * 256u + threadIdx.x;   // [0, 2*VPL)
    const unsigned isv = tid >> 21;                         // 0 = keys, 1 = values
    const unsigned v   = tid & (VPL - 1u);

    const unsigned b  = v >> 19;
    const unsigned s  = (v >> 7) & (MAX_SEQ - 1u);
    const unsigned h  = (v >> 4) & (KV_HEADS - 1u);
    const unsigned d8 = v & (VPROW - 1u);

    const v4u* src = (isv ? vco : kco) + (layer_idx << 21) + v;
    v4u d = __builtin_nontemporal_load(src);

    const unsigned HQ   = KV_HEADS * (unsigned)n_rep;
    const unsigned base = (((b * MAX_SEQ + s) * HQ + h * (unsigned)n_rep) << 4) + d8;
    v4u* out = isv ? vals : keys;
#pragma unroll 4
    for (int rr = 0; rr < n_rep; ++rr) {
        __builtin_nontemporal_store(d, out + base + (rr << 4));
    }
}

extern "C" void kernel_launch(void* const* d_in, const int* in_sizes, int n_in,
                              void* d_out, int out_size, void* d_ws, size_t ws_size,
                              hipStream_t stream)
{
    (void)in_sizes; (void)n_in; (void)d_ws; (void)ws_size;

    // setup_inputs order: xk, xv, k_cache, v_cache, layer_idx, cur_pos, n_rep
    const v4u* xk      = (const v4u*)d_in[0];
    const v4u* xv      = (const v4u*)d_in[1];
    const v4u* kin     = (const v4u*)d_in[2];
    const v4u* vin     = (const v4u*)d_in[3];
    const int* p_layer = (const int*)d_in[4];
    const int* p_pos   = (const int*)d_in[5];

    // n_rep determines the output layout; derive it from out_size (host-side,
    // deterministic, no device reads -> graph-capture safe).
    const long long CACHE_E = (long long)LAYERS * BSZ * MAX_SEQ * KV_HEADS * HEAD_DIM; // 134217728
    const long long PER_REP = (long long)BSZ * MAX_SEQ * KV_HEADS * HEAD_DIM;          // 16777216
    const int n_rep = (int)(((long long)out_size - 2 * CACHE_E) / (2 * PER_REP));
    const long long keys_e = (long long)n_rep * PER_REP;

    // d_out layout: keys | values | k_cache | v_cache (flat, bf16)
    __hip_bfloat16* outb = (__hip_bfloat16*)d_out;
    v4u* keys = (v4u*)(outb);
    v4u* vals = (v4u*)(outb + keys_e);
    v4u* kout = (v4u*)(outb + 2 * keys_e);
    v4u* vout = (v4u*)(outb + 2 * keys_e + CACHE_E);

    // 1) full cache copy + fused slice insert: 2*VPC vectors, VPT per thread
    kv_cache_copy_insert<<<(2 * VPC / VPT) / 256, 256, 0, stream>>>(
        kin, vin, xk, xv, p_layer, p_pos, kout, vout);

    // 2) GQA expansion of the updated layer (reads kernel-1 output, stream-ordered)
    kv_gqa_expand<<<(2 * VPL) / 256, 256, 0, stream>>>(
        kout, vout, keys, vals, p_layer, n_rep);
}